// Conv2d_48653389529482
// MI455X (gfx1250) — compile-verified
//
#include <hip/hip_runtime.h>

typedef __attribute__((ext_vector_type(2))) float v2f;
typedef __attribute__((ext_vector_type(8))) float v8f;

#define CH   16   // in channels
#define OC   16   // out channels
#define HH   512
#define WW   512
#define ROWS 8    // output rows per wave

// y[n,o,i,j] = sum_{ki,kj,c} x[n,c,i+ki-1,j+kj-1] * Kmem[(c*9+ki*3+kj)*16+o]
// with y[:, :, 511, :] = y[:, :, :, 511] = 0 (reference loop-bound bug).
__global__ __launch_bounds__(256) void conv3x3_wmma_f32(
    const float* __restrict__ x, const float* __restrict__ Kw,
    float* __restrict__ y) {
  const int lane = threadIdx.x & 31;
  const int wave = threadIdx.x >> 5;

  int b = blockIdx.x;
  const int jt = b & 31;  b >>= 5;   // column tile 0..31
  const int rg = b & 7;   b >>= 3;   // row group   0..7
  const int nb = b;                  // batch       0..15

  const int i0 = rg * 64 + wave * ROWS;  // first output row for this wave
  const int j0 = jt * 16;

  const int mn = lane & 15;          // A: out-channel o ; B: pixel n
  const int kb = (lane >> 4) * 2;    // K base in chunk: lanes0-15 -> k=0,1 ; lanes16-31 -> k=2,3

  // ---- Preload all A (weight) fragments A[ki][kj][q], q = channel chunk ----
  v2f A[3][3][4];
#pragma unroll
  for (int ki = 0; ki < 3; ++ki)
#pragma unroll
    for (int kj = 0; kj < 3; ++kj)
#pragma unroll
      for (int q = 0; q < 4; ++q) {
        const int c0 = q * 4 + kb;
        A[ki][kj][q].x = Kw[((c0 + 0) * 9 + ki * 3 + kj) * 16 + mn];
        A[ki][kj][q].y = Kw[((c0 + 1) * 9 + ki * 3 + kj) * 16 + mn];
      }

  const float* xb = x + (size_t)nb * CH * HH * WW;

  // Column handling, invariant across rows: clamped address + zero mask.
  // kj==1 -> col = j0+mn in [0,511] always: no mask needed.
  int  colc[3];
  bool colok[3];
#pragma unroll
  for (int kj = 0; kj < 3; ++kj) {
    const int col = j0 + mn + kj - 1;
    colok[kj] = (unsigned)col < (unsigned)WW;
    colc[kj]  = col < 0 ? 0 : (col > WW - 1 ? WW - 1 : col);
  }

  // ---- B ring buffer: 3 input rows x 4 channel chunks x 3 kj shifts ----
  v2f Bf[3][4][3];

  auto load_row = [&](int s, int r) {
    if ((unsigned)r < (unsigned)HH) {   // wave-uniform guard, hot path
#pragma unroll
      for (int q = 0; q < 4; ++q) {
        const int c0 = q * 4 + kb;
        const float* p = xb + ((size_t)c0 * HH + r) * WW;
#pragma unroll
        for (int kj = 0; kj < 3; ++kj) {
          const float vx = p[colc[kj]];
          const float vy = p[(size_t)HH * WW + colc[kj]];  // channel c0+1
          if (kj == 1) {                 // provably in range
            Bf[s][q][kj].x = vx;
            Bf[s][q][kj].y = vy;
          } else {
            Bf[s][q][kj].x = colok[kj] ? vx : 0.0f;
            Bf[s][q][kj].y = colok[kj] ? vy : 0.0f;
          }
        }
      }
    } else {                             // cold path: padding rows only
#pragma unroll
      for (int q = 0; q < 4; ++q)
#pragma unroll
        for (int kj = 0; kj < 3; ++kj)
          Bf[s][q][kj] = (v2f){0.0f, 0.0f};
    }
  };

  // slot(r) = (r - i0 + 1) mod 3 ; all indices static after unroll
  load_row(0, i0 - 1);
  load_row(1, i0);

#pragma unroll
  for (int ii = 0; ii < ROWS; ++ii) {
    load_row((ii + 2) % 3, i0 + ii + 1);   // bring in row i+1

    v8f acc = {0.f, 0.f, 0.f, 0.f, 0.f, 0.f, 0.f, 0.f};
#pragma unroll
    for (int ki = 0; ki < 3; ++ki) {
      const int s = (ii + ki) % 3;         // ring slot for input row i-1+ki
#pragma unroll
      for (int kj = 0; kj < 3; ++kj)
#pragma unroll
        for (int q = 0; q < 4; ++q)
          acc = __builtin_amdgcn_wmma_f32_16x16x4_f32(
              false, A[ki][kj][q], false, Bf[s][q][kj],
              (short)0, acc, false, false);
    }

    // ---- store 16x16 tile (nontemporal: y is write-once, keep L2 for x) ----
    const int i = i0 + ii;
    const int j = j0 + mn;
    const int mbase = (lane >> 4) * 8;     // C/D layout: vgpr v -> M = mbase + v
    const bool live = (i < HH - 1) && (j < WW - 1);
    float* yb = y + (((size_t)nb * OC + mbase) * HH + i) * WW + j;
#pragma unroll
    for (int v = 0; v < 8; ++v) {
      const float val = live ? acc[v] : 0.0f;
      __builtin_nontemporal_store(val, &yb[(size_t)v * HH * WW]);
    }
  }
}

extern "C" void kernel_launch(void* const* d_in, const int* in_sizes, int n_in,
                              void* d_out, int out_size, void* d_ws, size_t ws_size,
                              hipStream_t stream) {
  const float* x  = (const float*)d_in[0];   // (16,16,512,512) f32
  const float* Kw = (const float*)d_in[1];   // (3,3,16,16) f32, read as flat
  float* y = (float*)d_out;                  // (16,16,512,512) f32
  (void)in_sizes; (void)n_in; (void)out_size; (void)d_ws; (void)ws_size;

  // 16 batches * 8 row groups * 32 column tiles = 4096 blocks, 8 waves each
  conv3x3_wmma_f32<<<dim3(16 * 8 * 32), dim3(256), 0, stream>>>(x, Kw, y);
}